// Loss_net_58110907515037
// MI455X (gfx1250) — compile-verified
//
#include <hip/hip_runtime.h>

typedef float v2f __attribute__((ext_vector_type(2)));
typedef float v8f __attribute__((ext_vector_type(8)));

#define NI   11      // M+1 basis functions
#define MB   10      // M
#define LB   3
#define HIDN 5
#define DD   3
#define NSTEP 10
#define HH   0.1f

struct F3 { float x, y, z; };
struct EV { float vx, vy, vz, dv; };

// All tables live in LDS, laid out directly in WMMA operand register order.
struct Tables {
  float2 Bv[NI][32];        // B operand (K x N): per-lane {K0orK2, K1orK3} of W1-stack
  float  B1[NI][32];        // C operand broadcast: b1 per N-column
  float  W2t[NI][DD][16];   // stacked W2 rows (k x (l*5+j)), padded to 16
  float  Ssum[NI][16];      // S[i][n] = sum_k W2[i,l,k,j]*W1[i,l,j,k]
  float  B2s[NI][DD];       // sum_l b2[i,l,k]
  float  T[2][32][16];      // per-wave transpose scratch: [activeI][sample][unit]
};

__device__ __forceinline__ float fast_tanh(float x) {
  // tanh(x) = 1 - 2/(exp(2x)+1); v_exp_f32 + v_rcp_f32 (fast HW rcp, ~1 ulp)
  float xc = fminf(fmaxf(x, -15.f), 15.f);
  float e  = __expf(2.f * xc);
  return __builtin_fmaf(-2.f, __builtin_amdgcn_rcpf(e + 1.f), 1.f);
}

// Wave-cooperative field eval: 32 samples (one per lane), returns v (and div).
// Layer 1 (x -> 15 hidden pre-acts, for both active hat-basis indices) runs on
// V_WMMA_F32_16X16X4_F32; layer 2 (15 -> 3) + tanh + divergence are per-lane VALU.
__device__ __forceinline__ EV evalV(Tables* tb, float x0, float x1, float x2,
                                    float t, bool needDiv)
{
  const int  lane = threadIdx.x;
  const int  nn   = lane & 15;
  const bool hi   = lane >= 16;

  // Hat basis: only i0 and i0+1 are active.
  float u  = t * (float)MB;
  int   i0 = (int)floorf(u);
  i0 = i0 < 0 ? 0 : (i0 > MB - 1 ? MB - 1 : i0);
  float frac = u - (float)i0;

  __syncthreads();  // protect T against previous eval's readers (wave-barrier cheap, 1 wave/WG)

#pragma unroll
  for (int b = 0; b < 2; ++b) {
    const int i = i0 + b;
    float2 bvp = tb->Bv[i][lane];
    v2f B; B[0] = bvp.x; B[1] = bvp.y;
    float c = tb->B1[i][lane];
    v8f C = {c, c, c, c, c, c, c, c};
#pragma unroll
    for (int tile = 0; tile < 2; ++tile) {
      // Build A (16 samples x K=4, K=3 used): lanes 0-15 carry K0,K1; 16-31 carry K2,K3.
      int   src = nn + tile * 16;
      float sx0 = __shfl(x0, src, 32);
      float sx1 = __shfl(x1, src, 32);
      float sx2 = __shfl(x2, src, 32);
      v2f A; A[0] = hi ? sx2 : sx0; A[1] = hi ? 0.f : sx1;
      v8f Dt = __builtin_amdgcn_wmma_f32_16x16x4_f32(
          false, A, false, B, (short)0, C, false, false);
      // D layout: VGPR v = {lanes0-15: (M=v, N=lane)}, {lanes16-31: (M=v+8, N=lane-16)}
#pragma unroll
      for (int v = 0; v < 8; ++v) {
        int smp = (hi ? v + 8 : v) + tile * 16;
        tb->T[b][smp][nn] = Dt[v];
      }
    }
  }
  __syncthreads();

  EV r; r.vx = 0.f; r.vy = 0.f; r.vz = 0.f; r.dv = 0.f;
#pragma unroll
  for (int b = 0; b < 2; ++b) {
    const int   i  = i0 + b;
    const float pb = b ? frac : (1.f - frac);
    float a[15];
#pragma unroll
    for (int c2 = 0; c2 < 15; ++c2) a[c2] = fast_tanh(tb->T[b][lane][c2]);
    float acc[3];
#pragma unroll
    for (int k = 0; k < 3; ++k) {
      float s = tb->B2s[i][k];
#pragma unroll
      for (int c2 = 0; c2 < 15; ++c2)
        s = __builtin_fmaf(tb->W2t[i][k][c2], a[c2], s);
      acc[k] = s;
    }
    r.vx = __builtin_fmaf(pb, acc[0], r.vx);
    r.vy = __builtin_fmaf(pb, acc[1], r.vy);
    r.vz = __builtin_fmaf(pb, acc[2], r.vz);
    if (needDiv) {
      float d = 0.f;
#pragma unroll
      for (int c2 = 0; c2 < 15; ++c2)
        d = __builtin_fmaf(tb->Ssum[i][c2], 1.f - a[c2] * a[c2], d);
      r.dv = __builtin_fmaf(pb, d, r.dv);
    }
  }
  return r;
}

// One RK4 step of size H/4, with K1 supplied (shared with div/loss evals).
__device__ __forceinline__ F3 rk4x(Tables* tb, F3 X, float t, F3 K1) {
  const float s = HH * 0.25f;
  EV e = evalV(tb, X.x + 0.5f*s*K1.x, X.y + 0.5f*s*K1.y, X.z + 0.5f*s*K1.z, t + 0.5f*s, false);
  F3 K2 = {e.vx, e.vy, e.vz};
  e = evalV(tb, X.x + 0.5f*s*K2.x, X.y + 0.5f*s*K2.y, X.z + 0.5f*s*K2.z, t + 0.5f*s, false);
  F3 K3 = {e.vx, e.vy, e.vz};
  e = evalV(tb, X.x + s*K3.x, X.y + s*K3.y, X.z + s*K3.z, t + s, false);
  F3 K4 = {e.vx, e.vy, e.vz};
  const float c6 = s / 6.f;
  F3 Xn;
  Xn.x = X.x + c6 * (K1.x + 2.f*K2.x + 2.f*K3.x + K4.x);
  Xn.y = X.y + c6 * (K1.y + 2.f*K2.y + 2.f*K3.y + K4.y);
  Xn.z = X.z + c6 * (K1.z + 2.f*K2.z + 2.f*K3.z + K4.z);
  return Xn;
}

__global__ __launch_bounds__(32) void ode_main(
    const float* __restrict__ xin, const float* __restrict__ W1,
    const float* __restrict__ b1,  const float* __restrict__ W2,
    const float* __restrict__ b2,  float* __restrict__ ws, int Rtot)
{
  __shared__ Tables tb;
  const int  lane = threadIdx.x;
  const int  nn   = lane & 15;
  const bool hi   = lane >= 16;

  // ---- Build weight tables in WMMA operand layouts (once per block) ----
  for (int i = 0; i < NI; ++i) {
    float bv0 = 0.f, bv1 = 0.f, b1v = 0.f;
    if (nn < 15) {
      int l = nn / HIDN, j = nn % HIDN;
      const float* w1p = W1 + (((i * LB) + l) * HIDN + j) * DD;
      bv0 = hi ? w1p[2] : w1p[0];
      bv1 = hi ? 0.f    : w1p[1];
      b1v = b1[(i * LB + l) * HIDN + j];
    }
    tb.Bv[i][lane] = make_float2(bv0, bv1);
    tb.B1[i][lane] = b1v;
  }
  for (int idx = lane; idx < NI * 16; idx += 32) {
    int i = idx >> 4, n2 = idx & 15;
    float w2k[3] = {0.f, 0.f, 0.f};
    float ssv = 0.f;
    if (n2 < 15) {
      int l = n2 / HIDN, j = n2 % HIDN;
      for (int k = 0; k < DD; ++k) {
        float w2v = W2[(((i * LB) + l) * DD + k) * HIDN + j];
        w2k[k] = w2v;
        ssv += w2v * W1[(((i * LB) + l) * HIDN + j) * DD + k];  // S = diag(W2 W1)
      }
    }
    for (int k = 0; k < DD; ++k) tb.W2t[i][k][n2] = w2k[k];
    tb.Ssum[i][n2] = ssv;
  }
  for (int idx = lane; idx < NI * DD; idx += 32) {
    int i = idx / DD, k = idx % DD;
    tb.B2s[i][k] = b2[(i*LB + 0)*DD + k] + b2[(i*LB + 1)*DD + k] + b2[(i*LB + 2)*DD + k];
  }
  __syncthreads();

  int gid = blockIdx.x * 32 + lane;
  int smp = gid < Rtot ? gid : Rtot - 1;   // keep EXEC full for WMMA; mask at reduction
  const float LN2PI = 1.8378770664093453f;
  F3 X; X.x = xin[smp*3+0]; X.y = xin[smp*3+1]; X.z = xin[smp*3+2];
  float lnRo = -0.5f * (3.f*LN2PI + X.x*X.x + X.y*X.y + X.z*X.z);
  float s1 = 0.f;

  EV e0 = evalV(&tb, X.x, X.y, X.z, 0.f, true);
  for (int k = 0; k < NSTEP; ++k) {
    float tn = HH * (float)k;
    F3 K1 = {e0.vx, e0.vy, e0.vz};
    F3 X1 = rk4x(&tb, X, tn, K1);
    EV e1 = evalV(&tb, X1.x, X1.y, X1.z, tn + 0.25f*HH, true);
    F3 k1b = {e1.vx, e1.vy, e1.vz};
    F3 X2 = rk4x(&tb, X1, tn + 0.25f*HH, k1b);
    EV e2 = evalV(&tb, X2.x, X2.y, X2.z, tn + 0.5f*HH, true);
    F3 k1c = {e2.vx, e2.vy, e2.vz};
    F3 X3 = rk4x(&tb, X2, tn + 0.5f*HH, k1c);
    EV e3 = evalV(&tb, X3.x, X3.y, X3.z, tn + 0.75f*HH, true);
    F3 k1d = {e3.vx, e3.vy, e3.vz};
    F3 X4 = rk4x(&tb, X3, tn + 0.75f*HH, k1d);
    EV e4 = evalV(&tb, X4.x, X4.y, X4.z, tn + HH, true);

    const float c12 = HH / 12.f;                      // (H/2)/6
    lnRo += c12 * (-e0.dv - 4.f*e1.dv - e2.dv);
    lnRo += c12 * (-e2.dv - 4.f*e3.dv - e4.dv);
    s1 += (e0.vx*e0.vx + e0.vy*e0.vy + e0.vz*e0.vz)
        + 4.f*(e2.vx*e2.vx + e2.vy*e2.vy + e2.vz*e2.vz)
        + (e4.vx*e4.vx + e4.vy*e4.vy + e4.vz*e4.vz);
    X = X4; e0 = e4;
  }

  float dx = X.x - 1.f, dy = X.y - 1.f, dz = X.z - 1.f;
  float kl = lnRo + 0.5f * (3.f*LN2PI + dx*dx + dy*dy + dz*dz);
  float valid = (gid < Rtot) ? 1.f : 0.f;
  s1 *= valid; kl *= valid;
  for (int off = 16; off > 0; off >>= 1) {
    s1 += __shfl_xor(s1, off, 32);
    kl += __shfl_xor(kl, off, 32);
  }
  if (lane == 0) { ws[blockIdx.x*2 + 0] = s1; ws[blockIdx.x*2 + 1] = kl; }
}

__global__ void ode_finalize(const float* __restrict__ ws, float* __restrict__ out,
                             int nblk, int Rtot, int out_n)
{
  if (blockIdx.x == 0 && threadIdx.x == 0) {
    float s1 = 0.f, kl = 0.f;
    for (int b = 0; b < nblk; ++b) { s1 += ws[2*b]; kl += ws[2*b + 1]; }
    float loss1  = (HH / 6.f) * s1 / (float)Rtot;
    float lossKL = kl / (float)Rtot;          // LAM_KL = 1
    if (out_n > 0) out[0] = loss1 + lossKL;   // loss_F == 0 (Q == 0)
    if (out_n > 1) out[1] = loss1;
    if (out_n > 2) out[2] = lossKL;
    if (out_n > 3) out[3] = 0.f;
  }
}

extern "C" void kernel_launch(void* const* d_in, const int* in_sizes, int n_in,
                              void* d_out, int out_size, void* d_ws, size_t ws_size,
                              hipStream_t stream)
{
  (void)n_in; (void)ws_size;
  const float* xin = (const float*)d_in[0];
  const float* W1  = (const float*)d_in[1];
  const float* b1  = (const float*)d_in[2];
  const float* W2  = (const float*)d_in[3];
  const float* b2  = (const float*)d_in[4];
  int Rtot = in_sizes[0] / 3;
  int nblk = (Rtot + 31) / 32;
  ode_main<<<nblk, 32, 0, stream>>>(xin, W1, b1, W2, b2, (float*)d_ws, Rtot);
  ode_finalize<<<1, 1, 0, stream>>>((const float*)d_ws, (float*)d_out, nblk, Rtot, out_size);
}